// JPEGEncoder_87368224735247
// MI455X (gfx1250) — compile-verified
//
#include <hip/hip_runtime.h>
#include <math.h>

// ---------------- problem constants ----------------
#define BB 32
#define HH 512
#define WW 512
#define W8 64
#define NB 4096            // 64*64 blocks per image

typedef __attribute__((ext_vector_type(2))) float v2f;
typedef __attribute__((ext_vector_type(8))) float v8f;

// output layout (floats): qdct | blocks | enc_qf | enc_scale | y_qf_min | y_qf_max | c_qf_min | c_qf_max
static const size_t N_QDCT    = (size_t)BB * 3 * NB * 64;     // 25,165,824
static const size_t OFF_BLOCKS= N_QDCT;
static const size_t OFF_QF    = 2 * N_QDCT;                   // 50,331,648
static const size_t N_MAP     = (size_t)BB * HH * WW * 2;     // 16,777,216
static const size_t OFF_SC    = OFF_QF + N_MAP;
static const size_t OFF_MM    = OFF_SC + N_MAP;               // 83,886,080

// workspace layout (floats)
static const size_t WS_E  = 0;                                // [B][3][NB] per-channel log1p AC energy
static const size_t WS_QF = (size_t)BB * 3 * NB;              // [B][2][NB] per-block QF
static const size_t WS_SC = WS_QF + (size_t)BB * 2 * NB;      // [B][2][NB] per-block scale

__constant__ float kYQ[64] = {
  16,11,10,16,24,40,51,61, 12,12,14,19,26,58,60,55, 14,13,16,24,40,57,69,56,
  14,17,22,29,51,87,80,62, 18,22,37,56,68,109,103,77, 24,36,55,64,81,104,113,92,
  49,64,78,87,103,121,120,101, 72,92,95,98,112,100,103,99};
__constant__ float kCQ[64] = {
  17,18,24,47,99,99,99,99, 18,21,26,66,99,99,99,99, 24,26,56,99,99,99,99,99,
  47,66,99,99,99,99,99,99, 99,99,99,99,99,99,99,99, 99,99,99,99,99,99,99,99,
  99,99,99,99,99,99,99,99, 99,99,99,99,99,99,99,99};
__constant__ int kZZ[64] = {
  0,1,8,16,9,2,3,10,17,24,32,25,18,11,4,5,12,19,26,33,40,48,41,34,27,20,13,6,
  7,14,21,28,35,42,49,56,57,50,43,36,29,22,15,23,30,37,44,51,58,59,52,45,38,31,
  39,46,53,60,61,54,47,55,62,63};
// rows of RGB->YCC matrix + offset, per channel
__constant__ float kYCC[12] = {
   0.299f,     0.587f,    0.114f,    0.0f,
  -0.168736f, -0.331264f, 0.5f,      128.0f,
   0.5f,      -0.418688f,-0.081312f, 128.0f};

// Orthonormal 8x8 DCT-II matrix, precomputed to f64 precision (matches numpy f64 -> f32 cast).
// DCT_M[i][j] = 0.5*cos((2j+1)*i*pi/16); row 0 scaled by 1/sqrt(2).
#define D_A 0.35355339059327376220f  // 0.5/sqrt(2) and 0.5*cos(4pi/16)
#define D_B 0.49039264020161522457f  // 0.5*cos(1pi/16)
#define D_C 0.46193976625564337806f  // 0.5*cos(2pi/16)
#define D_D 0.41573480615127261854f  // 0.5*cos(3pi/16)
#define D_E 0.27778511650980111237f  // 0.5*cos(5pi/16)
#define D_F 0.19134171618254488586f  // 0.5*cos(6pi/16)
#define D_G 0.09754516100806413392f  // 0.5*cos(7pi/16)
__constant__ float kDCT[64] = {
   D_A,  D_A,  D_A,  D_A,  D_A,  D_A,  D_A,  D_A,
   D_B,  D_D,  D_E,  D_G, -D_G, -D_E, -D_D, -D_B,
   D_C,  D_F, -D_F, -D_C, -D_C, -D_F,  D_F,  D_C,
   D_D, -D_G, -D_B, -D_E,  D_E,  D_B,  D_G, -D_D,
   D_A, -D_A, -D_A,  D_A,  D_A, -D_A, -D_A,  D_A,
   D_E, -D_B,  D_G,  D_D, -D_D, -D_G,  D_B, -D_E,
   D_F, -D_C,  D_C, -D_F, -D_F,  D_C, -D_C,  D_F,
   D_G, -D_E,  D_D, -D_B,  D_B, -D_D,  D_E, -D_G};

// C2 = diag(DCT_M, DCT_M) entry (m,k) for 16x16 indices
__device__ __forceinline__ float c2val(int m, int k) {
  return (((m ^ k) & 8) != 0) ? 0.0f : kDCT[(m & 7) * 8 + (k & 7)];
}

// ---------------- kernel 1: YCC + 2x 16x16 WMMA DCT + AC energy ----------------
// grid (1024 tiles, 3 channels, B); block = 1 wave of 32.
// Wave owns a 16x16 pixel tile (= 2x2 JPEG blocks) of one channel.
// f32 WMMA layouts (ISA 7.12.2): A(16x4): vgpr=k&1, lane=(k>>1)*16+m.
// B(4x16): vgpr=k&1, lane=(k>>1)*16+n.  C/D(16x16): vgpr=m&7, lane=(m>>3)*16+n.
__global__ __launch_bounds__(32) void k_dct(const float* __restrict__ rgb,
                                            float* __restrict__ qdct,
                                            float* __restrict__ ws_e) {
  __shared__ float lds[256];
  const int lane = threadIdx.x;
  const int hi = lane >> 4, lo = lane & 15;
  const int tX = blockIdx.x & 31, tY = blockIdx.x >> 5;   // tile coords (16px units)
  const int ch = blockIdx.y, b = blockIdx.z;
  const float c0 = kYCC[ch*4+0], c1 = kYCC[ch*4+1], c2 = kYCC[ch*4+2], co = kYCC[ch*4+3];

  // This lane's 8 DCT-matrix entries, shared by matmul-1 A and matmul-2 B operands:
  // cvals[kk*2+j] = C2[lo][4*kk + 2*hi + j]
  float cvals[8];
#pragma unroll
  for (int kk = 0; kk < 4; ++kk)
#pragma unroll
    for (int j = 0; j < 2; ++j)
      cvals[kk*2 + j] = c2val(lo, 4*kk + 2*hi + j);

  // ---- T = C2 @ X   (A = C2 cols chunk, B = X rows chunk; K=16 via 4x K=4) ----
  v8f t = {0.f,0.f,0.f,0.f,0.f,0.f,0.f,0.f};
#pragma unroll
  for (int kk = 0; kk < 4; ++kk) {
    v2f a, x;
#pragma unroll
    for (int j = 0; j < 2; ++j) {
      const int kidx = 4*kk + 2*hi + j;                    // K index within 0..15
      const int py = tY*16 + kidx, px = tX*16 + lo;        // X(k, n): row=k, col=n=lo
      const float* p = rgb + (((size_t)b*HH + py)*WW + px)*3;
      x[j] = rintf(p[0]*c0 + p[1]*c1 + p[2]*c2 + co) - 128.0f;   // hard_round YCC - 128
      a[j] = cvals[kk*2 + j];                              // A(m=lo, k)
    }
    t = __builtin_amdgcn_wmma_f32_16x16x4_f32(false, a, false, x, (short)0, t, false, false);
  }
  // restripe T (D-layout) -> A-layout through LDS
#pragma unroll
  for (int r = 0; r < 8; ++r) lds[(r + 8*hi)*16 + lo] = t[r];
  __syncthreads();

  // ---- D = T @ C2^T  (B(k,n) = C2[n][k]) ----
  v8f d = {0.f,0.f,0.f,0.f,0.f,0.f,0.f,0.f};
#pragma unroll
  for (int kk = 0; kk < 4; ++kk) {
    v2f a, bm;
#pragma unroll
    for (int j = 0; j < 2; ++j) {
      const int kidx = 4*kk + 2*hi + j;
      a[j]  = lds[lo*16 + kidx];                           // T(m=lo, k)
      bm[j] = cvals[kk*2 + j];                             // C2^T(k, n=lo) = C2[lo][k]
    }
    d = __builtin_amdgcn_wmma_f32_16x16x4_f32(false, a, false, bm, (short)0, d, false, false);
  }

  // lane holds D(r+8*hi, lo): one column of one 8x8 block
  const int bi = tY*2 + hi, bj = tX*2 + (lo >> 3);
  const int blk = bi*W8 + bj;
  float* op = qdct + ((((size_t)b*3 + ch)*NB + blk)*64 + (lo & 7));
  float s = 0.f;
#pragma unroll
  for (int r = 0; r < 8; ++r) { op[r*8] = d[r]; s += d[r]*d[r]; }
  if ((lo & 7) == 0) s -= d[0]*d[0];                       // exclude DC
  s += __shfl_xor(s, 1, 32);
  s += __shfl_xor(s, 2, 32);
  s += __shfl_xor(s, 4, 32);                               // 8 lanes per block
  if ((lo & 7) == 0)
    ws_e[((size_t)b*3 + ch)*NB + blk] = log1pf(s + 1e-6f);
}

// ---------------- kernel 2: per-image rank -> QF/scale + QF min/max ----------------
// grid (2, B): c=0 luma, c=1 chroma. Two-phase reduction inside one workgroup.
__global__ __launch_bounds__(256) void k_rank(const float* __restrict__ ws_e,
                                              float* __restrict__ ws_qf,
                                              float* __restrict__ ws_sc,
                                              float* __restrict__ out) {
  const int c = blockIdx.x, b = blockIdx.y, tid = threadIdx.x;
  __shared__ float red[16];
  __shared__ float bcast[2];
  const float* e0 = ws_e + (size_t)b*3*NB;

  float mn = 1e30f, mx = -1e30f;
  for (int i = tid; i < NB; i += 256) {
    float e = (c == 0) ? e0[i] : 0.5f*(e0[NB + i] + e0[2*NB + i]);
    mn = fminf(mn, e); mx = fmaxf(mx, e);
  }
#pragma unroll
  for (int o = 16; o > 0; o >>= 1) {
    mn = fminf(mn, __shfl_xor(mn, o, 32));
    mx = fmaxf(mx, __shfl_xor(mx, o, 32));
  }
  if ((tid & 31) == 0) { red[tid >> 5] = mn; red[8 + (tid >> 5)] = mx; }
  __syncthreads();
  if (tid == 0) {
    mn = red[0]; mx = red[8];
#pragma unroll
    for (int w = 1; w < 8; ++w) { mn = fminf(mn, red[w]); mx = fmaxf(mx, red[8 + w]); }
    bcast[0] = mn; bcast[1] = mx;
  }
  __syncthreads();
  const float emn = bcast[0];
  const float inv = 1.0f / (bcast[1] - bcast[0] + 1e-6f);

  float* qfp = ws_qf + ((size_t)b*2 + c)*NB;
  float* scp = ws_sc + ((size_t)b*2 + c)*NB;
  float qmn = 1e30f, qmx = -1e30f;
  for (int i = tid; i < NB; i += 256) {
    float e = (c == 0) ? e0[i] : 0.5f*(e0[NB + i] + e0[2*NB + i]);
    float rank = (e - emn) * inv;
    float qf = rintf(2.0f + rank * 48.0f);
    qf = fminf(fmaxf(qf, 2.0f), 50.0f);
    float sc = rintf((qf < 50.0f) ? 5000.0f / qf : 200.0f - 2.0f*qf) * 0.01f;
    qfp[i] = qf; scp[i] = sc;
    qmn = fminf(qmn, qf); qmx = fmaxf(qmx, qf);
  }
#pragma unroll
  for (int o = 16; o > 0; o >>= 1) {
    qmn = fminf(qmn, __shfl_xor(qmn, o, 32));
    qmx = fmaxf(qmx, __shfl_xor(qmx, o, 32));
  }
  __syncthreads();
  if ((tid & 31) == 0) { red[tid >> 5] = qmn; red[8 + (tid >> 5)] = qmx; }
  __syncthreads();
  if (tid == 0) {
    qmn = red[0]; qmx = red[8];
#pragma unroll
    for (int w = 1; w < 8; ++w) { qmn = fminf(qmn, red[w]); qmx = fmaxf(qmx, red[8 + w]); }
    float* mm = out + OFF_MM;
    if (c == 0) { mm[b] = qmn; mm[32 + b] = qmx; }
    else        { mm[64 + b] = qmn; mm[96 + b] = qmx; }
  }
}

// ---------------- kernel 3: quantize in place + QF/scale pixel maps ----------------
// grid (NB, B), block 64: one thread per coefficient of one 8x8 block (all 3 channels).
__global__ __launch_bounds__(64) void k_quant(float* __restrict__ qdct,
                                              float* __restrict__ out_qf,
                                              float* __restrict__ out_sc,
                                              const float* __restrict__ ws_qf,
                                              const float* __restrict__ ws_sc) {
  const int blk = blockIdx.x, b = blockIdx.y, t = threadIdx.x;
  const float qf_y = ws_qf[((size_t)b*2 + 0)*NB + blk];
  const float qf_c = ws_qf[((size_t)b*2 + 1)*NB + blk];
  const float s_y  = ws_sc[((size_t)b*2 + 0)*NB + blk];
  const float s_c  = ws_sc[((size_t)b*2 + 1)*NB + blk];

  const size_t base = (((size_t)b*3)*NB + blk)*64 + t;
  const size_t chs  = (size_t)NB * 64;
  qdct[base]         = rintf(qdct[base]         / (s_y * kYQ[t]));
  qdct[base +   chs] = rintf(qdct[base +   chs] / (s_c * kCQ[t]));
  qdct[base + 2*chs] = rintf(qdct[base + 2*chs] / (s_c * kCQ[t]));

  const int py = (blk >> 6)*8 + (t >> 3), px = (blk & 63)*8 + (t & 7);
  const size_t pidx = (((size_t)b*HH + py)*WW + px)*2;
  out_qf[pidx] = qf_y; out_qf[pidx + 1] = qf_c;
  out_sc[pidx] = s_y;  out_sc[pidx + 1] = s_c;
}

// ---------------- kernel 4: delta + zigzag + log2 ----------------
__global__ __launch_bounds__(256) void k_blocks(const float* __restrict__ qdct,
                                                float* __restrict__ out_blocks) {
  const size_t idx = (size_t)blockIdx.x * 256 + threadIdx.x;
  if (idx >= N_QDCT) return;
  const int k = (int)(idx & 63);
  const size_t row = idx >> 6;          // (b*3+ch)*NB + n
  const int n = (int)(row & (NB - 1));
  const int j = kZZ[k];
  const size_t basep = row * 64;
  const float cur = qdct[basep + j];
  const float prev = (n > 0) ? qdct[basep - 64 + j] : 0.0f;
  out_blocks[idx] = log2f(fabsf(cur - prev) + 1.0f);
}

// ---------------- launcher ----------------
extern "C" void kernel_launch(void* const* d_in, const int* in_sizes, int n_in,
                              void* d_out, int out_size, void* d_ws, size_t ws_size,
                              hipStream_t stream) {
  (void)in_sizes; (void)n_in; (void)out_size; (void)ws_size;
  const float* rgb = (const float*)d_in[0];
  float* out = (float*)d_out;
  float* ws  = (float*)d_ws;

  float* qdct   = out;
  float* blocks = out + OFF_BLOCKS;
  float* map_qf = out + OFF_QF;
  float* map_sc = out + OFF_SC;

  k_dct   <<<dim3(1024, 3, BB), dim3(32),  0, stream>>>(rgb, qdct, ws + WS_E);
  k_rank  <<<dim3(2, BB),       dim3(256), 0, stream>>>(ws + WS_E, ws + WS_QF, ws + WS_SC, out);
  k_quant <<<dim3(NB, BB),      dim3(64),  0, stream>>>(qdct, map_qf, map_sc, ws + WS_QF, ws + WS_SC);
  k_blocks<<<dim3((unsigned)((N_QDCT + 255) / 256)), dim3(256), 0, stream>>>(qdct, blocks);
}